// QuantumEntangledAttention_74500502716751
// MI455X (gfx1250) — compile-verified
//
#include <hip/hip_runtime.h>
#include <hip/hip_bf16.h>
#include <math.h>

typedef __attribute__((ext_vector_type(2))) float v2f;
typedef __attribute__((ext_vector_type(4))) float v4f;
typedef __attribute__((ext_vector_type(8))) float v8f;

static constexpr int Bq = 8;
static constexpr int Sq = 2048;
static constexpr int Dq = 512;
static constexpr int NSTATES = Bq * Sq;          // 16384
static constexpr int NCHUNK = 128;               // j-chunks per batch in wsum
static constexpr int ROWS_PER_CHUNK = Sq / NCHUNK;  // 16
static constexpr float BASE_C = 0.30901699437494742f;  // cos(0.4*pi)
static constexpr float BASE_S = 0.95105651629515357f;  // sin(0.4*pi)

__device__ __forceinline__ float wave_sum32(float v) {
#pragma unroll
  for (int off = 16; off; off >>= 1) v += __shfl_xor(v, off, 32);
  return v;
}

// ---------------------------------------------------------------------------
// Kernel A: per-(b,s) quantum score. One wave32 per state; statevector of
// 256 complex amps lives in registers (8 complex per lane); cross-lane qubit
// gates via wave32 shuffles. `key` is streamed once -> non-temporal loads.
// ---------------------------------------------------------------------------
__global__ __launch_bounds__(256) void score_kernel(
    const float* __restrict__ key, const float* __restrict__ Wk,
    const float* __restrict__ bk, float* __restrict__ score) {
  const int lane = threadIdx.x & 31;
  const int wid = (blockIdx.x << 3) + (threadIdx.x >> 5);  // state index
  if (wid >= NSTATES) return;
  const float* krow = key + (size_t)wid * Dq;

  // ---- features th[16] = key_row . Wk + bk (each lane: 16 strided d's) ----
  float4 p0 = {0.f, 0.f, 0.f, 0.f}, p1 = p0, p2 = p0, p3 = p0;
#pragma unroll
  for (int e = 0; e < 16; ++e) {
    const int d = lane + 32 * e;
    const float kd = __builtin_nontemporal_load(krow + d);
    const float4* wr = (const float4*)(Wk + d * 16);
    const float4 w0 = wr[0], w1 = wr[1], w2 = wr[2], w3 = wr[3];
    p0.x = fmaf(kd, w0.x, p0.x); p0.y = fmaf(kd, w0.y, p0.y);
    p0.z = fmaf(kd, w0.z, p0.z); p0.w = fmaf(kd, w0.w, p0.w);
    p1.x = fmaf(kd, w1.x, p1.x); p1.y = fmaf(kd, w1.y, p1.y);
    p1.z = fmaf(kd, w1.z, p1.z); p1.w = fmaf(kd, w1.w, p1.w);
    p2.x = fmaf(kd, w2.x, p2.x); p2.y = fmaf(kd, w2.y, p2.y);
    p2.z = fmaf(kd, w2.z, p2.z); p2.w = fmaf(kd, w2.w, p2.w);
    p3.x = fmaf(kd, w3.x, p3.x); p3.y = fmaf(kd, w3.y, p3.y);
    p3.z = fmaf(kd, w3.z, p3.z); p3.w = fmaf(kd, w3.w, p3.w);
  }
  float th[16];
  th[0] = wave_sum32(p0.x) + bk[0];   th[1] = wave_sum32(p0.y) + bk[1];
  th[2] = wave_sum32(p0.z) + bk[2];   th[3] = wave_sum32(p0.w) + bk[3];
  th[4] = wave_sum32(p1.x) + bk[4];   th[5] = wave_sum32(p1.y) + bk[5];
  th[6] = wave_sum32(p1.z) + bk[6];   th[7] = wave_sum32(p1.w) + bk[7];
  th[8] = wave_sum32(p2.x) + bk[8];   th[9] = wave_sum32(p2.y) + bk[9];
  th[10] = wave_sum32(p2.z) + bk[10]; th[11] = wave_sum32(p2.w) + bk[11];
  th[12] = wave_sum32(p3.x) + bk[12]; th[13] = wave_sum32(p3.y) + bk[13];
  th[14] = wave_sum32(p3.z) + bk[14]; th[15] = wave_sum32(p3.w) + bk[15];

  // ---- entangled base state (factors over the 4 pairs) ----
  float ar[8], ai[8];
#pragma unroll
  for (int k = 0; k < 8; ++k) {
    const int i = lane * 8 + k;
    float br = 0.25f, bi = 0.f;  // (1/sqrt2)^4
#pragma unroll
    for (int p = 0; p < 4; ++p) {
      const int b1 = (i >> (7 - 2 * p)) & 1;  // control qubit bit
      const int b2 = (i >> (6 - 2 * p)) & 1;  // target qubit bit
      if (b1 == 0 && b2 == 1) { br = 0.f; bi = 0.f; }
      else if (b1 == 1 && b2 == 0) { br *= BASE_C; bi *= BASE_C; }
      else if (b1 == 1 && b2 == 1) {  // multiply by (-i * s)
        const float t = br;
        br = bi * BASE_S;
        bi = -t * BASE_S;
      }
    }
    ar[k] = br; ai[k] = bi;
  }

  // ---- apply per-qubit gates G = Gz * Gx ----
#pragma unroll
  for (int q = 0; q < 8; ++q) {
    const float thx = th[2 * q], thz = th[2 * q + 1];
    float sx, cx; __sincosf(0.5f * thx, &sx, &cx);
    if (!(thx > 0.f)) { sx = 0.f; cx = 1.f; }
    float sz, cz; __sincosf(0.5f * thz, &sz, &cz);
    if (!(thz > 0.f)) { sz = 0.f; cz = 1.f; }
    // G00=(cz*cx,-sz*cx) G01=(-sz*sx,-cz*sx) G10=(sz*sx,-cz*sx) G11=(cz*cx,sz*cx)
    const float g00r = cz * cx,  g00i = -sz * cx;
    const float g01r = -sz * sx, g01i = -cz * sx;
    const float g10r = sz * sx,  g10i = -cz * sx;
    const float g11r = cz * cx,  g11i = sz * cx;

    if (q < 5) {  // qubit bit lives in lane index (bit 4-q)
      const int mask = 1 << (4 - q);
      const int myb = (lane >> (4 - q)) & 1;
#pragma unroll
      for (int k = 0; k < 8; ++k) {
        const float pr = __shfl_xor(ar[k], mask, 32);
        const float pi = __shfl_xor(ai[k], mask, 32);
        const float a0r = myb ? pr : ar[k], a0i = myb ? pi : ai[k];
        const float a1r = myb ? ar[k] : pr, a1i = myb ? ai[k] : pi;
        const float gar = myb ? g10r : g00r, gai = myb ? g10i : g00i;
        const float gbr = myb ? g11r : g01r, gbi = myb ? g11i : g01i;
        ar[k] = gar * a0r - gai * a0i + gbr * a1r - gbi * a1i;
        ai[k] = gar * a0i + gai * a0r + gbr * a1i + gbi * a1r;
      }
    } else {  // qubit bit lives in k (bit 7-q)
      const int stride = 1 << (7 - q);
#pragma unroll
      for (int k0 = 0; k0 < 8; ++k0) {
        if ((k0 & stride) == 0) {
          const int k1 = k0 | stride;
          const float a0r = ar[k0], a0i = ai[k0];
          const float a1r = ar[k1], a1i = ai[k1];
          ar[k0] = g00r * a0r - g00i * a0i + g01r * a1r - g01i * a1i;
          ai[k0] = g00r * a0i + g00i * a0r + g01r * a1i + g01i * a1r;
          ar[k1] = g10r * a0r - g10i * a0i + g11r * a1r - g11i * a1i;
          ai[k1] = g10r * a0i + g10i * a0r + g11r * a1i + g11i * a1r;
        }
      }
    }
  }

  // ---- score = (1/4) * sum_i prob_i * #pairs-with-equal-bits ----
  float acc = 0.f;
#pragma unroll
  for (int k = 0; k < 8; ++k) {
    const int i = lane * 8 + k;
    const int cnt = 4 - __popc((i ^ (i >> 1)) & 0x55);
    acc += (ar[k] * ar[k] + ai[k] * ai[k]) * (float)cnt;
  }
  acc = wave_sum32(acc);
  if (lane == 0) score[wid] = acc * 0.25f;
}

// ---------------------------------------------------------------------------
// Kernel B: softmax over S per batch (one block per batch).
// ---------------------------------------------------------------------------
__global__ __launch_bounds__(256) void softmax_kernel(
    const float* __restrict__ score, float* __restrict__ w) {
  const int b = blockIdx.x, tid = threadIdx.x;
  __shared__ float red[256];
  const float* s = score + b * Sq;
  float m = -INFINITY;
#pragma unroll
  for (int r = 0; r < 8; ++r) m = fmaxf(m, s[tid + 256 * r]);
  red[tid] = m; __syncthreads();
  for (int k = 128; k; k >>= 1) {
    if (tid < k) red[tid] = fmaxf(red[tid], red[tid + k]);
    __syncthreads();
  }
  m = red[0]; __syncthreads();
  float e[8], z = 0.f;
#pragma unroll
  for (int r = 0; r < 8; ++r) { e[r] = __expf(s[tid + 256 * r] - m); z += e[r]; }
  red[tid] = z; __syncthreads();
  for (int k = 128; k; k >>= 1) {
    if (tid < k) red[tid] += red[tid + k];
    __syncthreads();
  }
  const float inv = 1.f / red[0];
#pragma unroll
  for (int r = 0; r < 8; ++r) w[b * Sq + tid + 256 * r] = e[r] * inv;
}

// ---------------------------------------------------------------------------
// Kernel C: partial weighted sums of `value` (the main HBM stream).
// grid: 8 batches x 128 j-chunks (1024 blocks x 128 threads); each thread
// owns one float4 column and streams 16 rows with non-temporal b128 loads.
// ---------------------------------------------------------------------------
__global__ __launch_bounds__(128) void wsum_kernel(
    const float* __restrict__ value, const float* __restrict__ w,
    float* __restrict__ P) {
  const int bid = blockIdx.x;            // 0..1023
  const int b = bid >> 7, c = bid & 127; // batch, chunk
  const int tid = threadIdx.x;           // float4 column 0..127
  const float* vb = value + ((size_t)b * Sq + (size_t)c * ROWS_PER_CHUNK) * Dq;
  const float* wb = w + b * Sq + c * ROWS_PER_CHUNK;
  v4f acc = {0.f, 0.f, 0.f, 0.f};
#pragma unroll 4
  for (int j = 0; j < ROWS_PER_CHUNK; ++j) {
    const v4f* vr = (const v4f*)(vb + (size_t)j * Dq) + tid;
    if (j + 4 < ROWS_PER_CHUNK) __builtin_prefetch(vr + Dq, 0, 1);
    const v4f v = __builtin_nontemporal_load(vr);
    const float wj = wb[j];
    acc.x = fmaf(wj, v.x, acc.x);
    acc.y = fmaf(wj, v.y, acc.y);
    acc.z = fmaf(wj, v.z, acc.z);
    acc.w = fmaf(wj, v.w, acc.w);
  }
  ((v4f*)(P + (size_t)bid * Dq))[tid] = acc;
}

// ---------------------------------------------------------------------------
// Kernel C2: fixed-order reduce of partials; pad vbar to 16 rows for WMMA.
// ---------------------------------------------------------------------------
__global__ __launch_bounds__(512) void reduce_pad_kernel(
    const float* __restrict__ P, float* __restrict__ A) {
  const int b = blockIdx.x;  // 0..15
  const int d = threadIdx.x;
  float s = 0.f;
  if (b < Bq) {
#pragma unroll 8
    for (int c = 0; c < NCHUNK; ++c) s += P[(size_t)((b << 7) + c) * Dq + d];
  }
  A[b * Dq + d] = s;
}

// ---------------------------------------------------------------------------
// Kernel D: 16xN = A(16xK) @ W(KxN) + bias via V_WMMA_F32_16X16X4_F32.
// One wave per 16-column tile; K stepped by 4.
// A layout: lane%16 = M row; VGPR v, lanegroup g -> K = k0 + 2*g + v.
// B layout: lane%16 = N col; VGPR v, lanegroup g -> K = k0 + 2*g + v.
// D layout: VGPR v, lanegroup g -> M = g*8 + v, N = lane%16.
// ---------------------------------------------------------------------------
__global__ __launch_bounds__(32) void gemm16_wmma(
    const float* __restrict__ A, const float* __restrict__ W,
    const float* __restrict__ bias, float* __restrict__ out,
    int K, int N) {
  const int lane = threadIdx.x;
  const int n0 = blockIdx.x * 16;
  const int g = lane >> 4;
  const int r = lane & 15;
  v8f c = {};
  for (int k0 = 0; k0 < K; k0 += 4) {
    const int ka = k0 + 2 * g;
    v2f a, bm;
    a.x = A[r * K + ka];
    a.y = A[r * K + ka + 1];
    bm.x = W[(size_t)ka * N + n0 + r];
    bm.y = W[(size_t)(ka + 1) * N + n0 + r];
    c = __builtin_amdgcn_wmma_f32_16x16x4_f32(
        /*neg_a=*/false, a, /*neg_b=*/false, bm,
        /*c_mod=*/(short)0, c, /*reuse_a=*/false, /*reuse_b=*/false);
  }
  const float bb = bias[n0 + r];
#pragma unroll
  for (int v = 0; v < 8; ++v) {
    const int m = g * 8 + v;
    out[m * N + n0 + r] = c[v] + bb;
  }
}

// ---------------------------------------------------------------------------
// Kernel E: broadcast y[b,:] to all S rows. Output is written once and never
// re-read on-device -> non-temporal float4 stores.
// ---------------------------------------------------------------------------
__global__ __launch_bounds__(256) void bcast_kernel(
    const float* __restrict__ Y, float* __restrict__ out) {
  const size_t g = (size_t)blockIdx.x * 256 + threadIdx.x;
  const size_t e = g * 4;                  // element offset
  const int b = (int)(e >> 20);            // S*D = 2^20
  const int d = (int)(e & (size_t)(Dq - 1));
  const v4f y = *(const v4f*)(Y + b * Dq + d);
  __builtin_nontemporal_store(y, (v4f*)(out + e));
}

extern "C" void kernel_launch(void* const* d_in, const int* in_sizes, int n_in,
                              void* d_out, int out_size, void* d_ws, size_t ws_size,
                              hipStream_t stream) {
  // setup_inputs order: query, key, value, Wq, bq, Wk, bk, Wv, bv, Wo, bo
  const float* key   = (const float*)d_in[1];
  const float* value = (const float*)d_in[2];
  const float* Wk    = (const float*)d_in[5];
  const float* bk    = (const float*)d_in[6];
  const float* Wv    = (const float*)d_in[7];
  const float* bv    = (const float*)d_in[8];
  const float* Wo    = (const float*)d_in[9];
  const float* bo    = (const float*)d_in[10];
  float* out = (float*)d_out;

  float* ws = (float*)d_ws;
  float* score = ws;                   // 16384
  float* w     = ws + 16384;           // 16384
  float* P     = ws + 32768;           // 1024*512 = 524288
  float* A     = ws + 557056;          // 16*512
  float* T     = ws + 565248;          // 16*512
  float* Y     = ws + 573440;          // 16*512

  score_kernel<<<NSTATES / 8, 256, 0, stream>>>(key, Wk, bk, score);
  softmax_kernel<<<Bq, 256, 0, stream>>>(score, w);
  wsum_kernel<<<Bq * NCHUNK, 128, 0, stream>>>(value, w, P);
  reduce_pad_kernel<<<16, 512, 0, stream>>>(P, A);
  gemm16_wmma<<<Dq / 16, 32, 0, stream>>>(A, Wv, bv, T, Dq, Dq);
  gemm16_wmma<<<Dq / 16, 32, 0, stream>>>(T, Wo, bo, Y, Dq, Dq);
  bcast_kernel<<<(NSTATES * Dq) / 1024, 256, 0, stream>>>(Y, out);
}